// FullChipLitho_72103910965606
// MI455X (gfx1250) — compile-verified
//
#include <hip/hip_runtime.h>
#include <math.h>

// CDNA5 wave32 WMMA vector types
typedef float v2f __attribute__((ext_vector_type(2)));
typedef float v4f __attribute__((ext_vector_type(4)));
typedef float v8f __attribute__((ext_vector_type(8)));

#define KDIM   35
#define KPAD   36
#define NK     24
#define HH     1024
#define WW     1024
#define NIMG   2
#define PLANE  (HH*WW)            // 1048576
#define SZ1    (NIMG*PLANE)       // 2097152 (one output field)
#define PLANESZ (KPAD*WW*2)       // 73728 floats per (img,k): 18 pairs x 1024 x 4
// workspace layout (floats):
#define G_OFF    0                        // G[n][v][y] complex : 2*35*1024*2
#define G_SZ     (NIMG*KDIM*WW*2)         // 143360
#define SPEC_OFF (G_OFF + G_SZ)           // spec[n][u][v] complex : 2*35*35*2
#define A_OFF    148480                   // A[img(4)][k(24)][pair(18)][xp(1024)][4]
// total ws use: 148480 + 4*24*73728 = 7,226,368 floats (~29 MB)

__device__ __forceinline__ float sigmoidf_(float z) { return 1.0f / (1.0f + __expf(-z)); }

__device__ __forceinline__ void build_wtab(float* Wt, int tid, int nthr) {
    const float step = 6.28318530717958647692f / 1024.0f;
    for (int i = tid; i < 1024; i += nthr) {
        float s, c;
        __sincosf(step * (float)i, &s, &c);
        Wt[2*i]   = c;
        Wt[2*i+1] = s;
    }
}

// ---------------------------------------------------------------------------
// Kernel 1: fused 3x3 avg-pool + sigmoid + row-wise partial DFT (x -> v)
//   G[n][v][y] = sum_x s[n,y,x] * e^{-2*pi*i*(v-17)*x/1024}
// one block per (n,y)
// ---------------------------------------------------------------------------
__global__ void __launch_bounds__(128) row_dft_kernel(const float* __restrict__ mask,
                                                      float* __restrict__ ws)
{
    __shared__ float Wt[2048];
    __shared__ float cs[1026];
    __shared__ float srow[1024];
    const int tid = threadIdx.x;
    const int bid = blockIdx.x;          // 0..2047
    const int n = bid >> 10;
    const int y = bid & 1023;

    build_wtab(Wt, tid, 128);

    const float* m = mask + (size_t)n * PLANE;
    for (int x = tid; x < 1024; x += 128) {
        float v = m[y*1024 + x];
        if (y > 0)    v += m[(y-1)*1024 + x];
        if (y < 1023) v += m[(y+1)*1024 + x];
        cs[x+1] = v;
    }
    if (tid == 0) { cs[0] = 0.0f; cs[1025] = 0.0f; }
    __syncthreads();
    for (int x = tid; x < 1024; x += 128) {
        float p = (cs[x] + cs[x+1] + cs[x+2]) * (1.0f/9.0f);
        srow[x] = sigmoidf_(4.0f * (p - 0.5f));
    }
    __syncthreads();

    if (tid < KDIM) {
        const int f = (tid - 17 + 1024) & 1023;   // frequency (v-17) mod 1024
        float re = 0.0f, im = 0.0f;
        for (int x = 0; x < 1024; ++x) {
            int j = (f * x) & 1023;
            float c = Wt[2*j], s = Wt[2*j+1];
            float v = srow[x];
            re += v * c;     // e^{-i t} = (cos, -sin)
            im -= v * s;
        }
        float* g = ws + G_OFF + ((size_t)(n*KDIM + tid) * 1024 + y) * 2;
        g[0] = re; g[1] = im;
    }
}

// ---------------------------------------------------------------------------
// Kernel 2: column partial DFT (y -> u): spec[n][u][v] = sum_y G[n][v][y]*e^{-i...}
// one wave per output element (2*35*35 = 2450)
// ---------------------------------------------------------------------------
__global__ void __launch_bounds__(256) col_dft_kernel(float* __restrict__ ws)
{
    __shared__ float Wt[2048];
    const int tid = threadIdx.x;
    build_wtab(Wt, tid, 256);
    __syncthreads();

    const int widx = blockIdx.x * 8 + (tid >> 5);
    const int lane = tid & 31;
    if (widx >= NIMG*KDIM*KDIM) return;
    const int n = widx / (KDIM*KDIM);
    const int rem = widx % (KDIM*KDIM);
    const int u = rem / KDIM, v = rem % KDIM;
    const int f = (u - 17 + 1024) & 1023;

    const float* g = ws + G_OFF + (size_t)(n*KDIM + v) * 2048;
    float re = 0.0f, im = 0.0f;
    for (int y = lane; y < 1024; y += 32) {
        int j = (f * y) & 1023;
        float c = Wt[2*j], s = Wt[2*j+1];
        float gr = g[2*y], gi = g[2*y+1];
        re += gr*c + gi*s;      // (gr+i gi)*(c - i s)
        im += gi*c - gr*s;
    }
    for (int off = 16; off > 0; off >>= 1) {
        re += __shfl_down(re, off, 32);
        im += __shfl_down(im, off, 32);
    }
    if (lane == 0) {
        float* sp = ws + SPEC_OFF + (size_t)widx * 2;
        sp[0] = re; sp[1] = im;
    }
}

// ---------------------------------------------------------------------------
// Kernel 3: T = spec .* kernel / 2^20, then inverse stage-1 (v -> xp):
//   A[(u,xp)] = sum_v T[u][v] * e^{+2*pi*i*v*xp/1024}
// Stored in WMMA-fragment-native layout: [img][k][pair=u>>1][xp][4] where the
// 4 floats are {Ar[2p], Ar[2p+1], Ai[2p], Ai[2p+1]} at column xp -> one b128
// load per lane per K-step in the GEMM. u==35 is the zero pad row.
// one block per (img,k,u)
// ---------------------------------------------------------------------------
__global__ void __launch_bounds__(256) inv_stage1_kernel(const float* __restrict__ kfr,
                                                         const float* __restrict__ kfi,
                                                         const float* __restrict__ kdr,
                                                         const float* __restrict__ kdi,
                                                         float* __restrict__ ws)
{
    __shared__ float Wt[2048];
    __shared__ float Tr[KDIM], Ti[KDIM];
    const int tid = threadIdx.x;
    build_wtab(Wt, tid, 256);

    const int bid = blockIdx.x;            // 4*24*36
    const int u   = bid % KPAD;
    const int k   = (bid / KPAD) % NK;
    const int img = bid / (KPAD * NK);
    const int n = img >> 1, set = img & 1;
    const float* kr = set ? kdr : kfr;
    const float* ki = set ? kdi : kfi;

    if (tid < KDIM) {
        float tr = 0.0f, ti = 0.0f;
        if (u < KDIM) {
            const float* sp = ws + SPEC_OFF + ((size_t)(n*KDIM + u)*KDIM + tid)*2;
            float sr = sp[0], si = sp[1];
            float krv = kr[(size_t)(k*KDIM + u)*KDIM + tid];
            float kiv = ki[(size_t)(k*KDIM + u)*KDIM + tid];
            const float inv = 1.0f / 1048576.0f;   // ifft2 1/(H*W)
            tr = (sr*krv - si*kiv) * inv;
            ti = (sr*kiv + si*krv) * inv;
        }
        Tr[tid] = tr; Ti[tid] = ti;
    }
    __syncthreads();

    float* Abase = ws + A_OFF + (size_t)(img*NK + k) * PLANESZ
                 + (size_t)(u >> 1) * 4096 + (u & 1);
    for (int jj = 0; jj < 4; ++jj) {
        int xp = jj*256 + tid;
        float ar = 0.0f, ai = 0.0f;
        for (int v = 0; v < KDIM; ++v) {
            int j = (v * xp) & 1023;
            float c = Wt[2*j], s = Wt[2*j+1];
            float tr = Tr[v], ti = Ti[v];
            ar += tr*c - ti*s;      // e^{+i t}
            ai += tr*s + ti*c;
        }
        Abase[(size_t)xp*4]     = ar;
        Abase[(size_t)xp*4 + 2] = ai;
    }
}

// ---------------------------------------------------------------------------
// Kernel 4: WMMA inverse stage-2 (u -> y) + intensity accumulation over k.
// Complex (1024x36)x(36x1024) per (img,k) realified to K=72 fp32 GEMM using
// V_WMMA_F32_16X16X4_F32. One wave per PAIR of 16x16 output tiles stacked in
// y (B fragments are y-independent -> each b128 B-load feeds 8 WMMAs).
// Writes I_f into x_out slot and 0.9604*I_d into x_out_min slot of d_out.
// ---------------------------------------------------------------------------
__global__ void __launch_bounds__(256) wmma_intensity_kernel(const float* __restrict__ ws,
                                                             const float* __restrict__ sf,
                                                             const float* __restrict__ sd,
                                                             float* __restrict__ out)
{
    __shared__ float Wt[2048];
    const int tid = threadIdx.x;
    build_wtab(Wt, tid, 256);
    __syncthreads();

    const int wid  = blockIdx.x * 8 + (tid >> 5);   // 0..8191 tile-pairs
    const int lane = tid & 31;
    const int half = lane >> 4;
    const int q    = lane & 15;
    const int img  = wid >> 11;                     // (n,set)
    const int rem  = wid & 2047;
    const int ybase0 = (rem >> 6) << 5;             // 32-row slab
    const int ybase1 = ybase0 + 16;
    const int xbase  = (rem & 63) << 4;
    const int n = img >> 1, set = img & 1;
    const int y0 = ybase0 + q;
    const int y1 = ybase1 + q;
    const int col = xbase + q;

    // LHS fragments: L[y][j], j<36 = cos(2*pi*u*y/1024), j>=36 = sin(...)
    // A-matrix layout: lane (half,q): a.x -> K=2*half, a.y -> K=2*half+1 (per 4-block)
    v2f Lc0[9], Ls0[9], Lc1[9], Ls1[9];
#pragma unroll
    for (int kk = 0; kk < 9; ++kk) {
        int u0 = 4*kk + 2*half;
        int i00 = (u0 * y0) & 1023, i01 = ((u0 + 1) * y0) & 1023;
        int i10 = (u0 * y1) & 1023, i11 = ((u0 + 1) * y1) & 1023;
        Lc0[kk].x = Wt[2*i00];     Lc0[kk].y = Wt[2*i01];
        Ls0[kk].x = Wt[2*i00 + 1]; Ls0[kk].y = Wt[2*i01 + 1];
        Lc1[kk].x = Wt[2*i10];     Lc1[kk].y = Wt[2*i11];
        Ls1[kk].x = Wt[2*i10 + 1]; Ls1[kk].y = Wt[2*i11 + 1];
    }

    const float* scale = set ? sd : sf;
    float Iacc0[8] = {0.f,0.f,0.f,0.f,0.f,0.f,0.f,0.f};
    float Iacc1[8] = {0.f,0.f,0.f,0.f,0.f,0.f,0.f,0.f};

    for (int k = 0; k < NK; ++k) {
        // lane's B data for K-step kk lives at pair p = 2*kk + half, column col
        const v4f* Ab = (const v4f*)(ws + A_OFF + (size_t)(img*NK + k) * PLANESZ);
        v8f cre0 = {0.f,0.f,0.f,0.f,0.f,0.f,0.f,0.f};
        v8f cim0 = {0.f,0.f,0.f,0.f,0.f,0.f,0.f,0.f};
        v8f cre1 = {0.f,0.f,0.f,0.f,0.f,0.f,0.f,0.f};
        v8f cim1 = {0.f,0.f,0.f,0.f,0.f,0.f,0.f,0.f};
#pragma unroll
        for (int kk = 0; kk < 9; ++kk) {
            v4f b = Ab[(size_t)(2*kk + half) * 1024 + col];  // {Ar0,Ar1,Ai0,Ai1}
            v2f br;  br.x  = b.x;  br.y  = b.y;
            v2f bi;  bi.x  = b.z;  bi.y  = b.w;
            v2f nbi; nbi.x = -b.z; nbi.y = -b.w;
            // xr = Er*Ar - Ei*Ai ; xi = Er*Ai + Ei*Ar   (D = A*B + C)
            cre0 = __builtin_amdgcn_wmma_f32_16x16x4_f32(false, Lc0[kk], false, br,  (short)0, cre0, false, false);
            cre0 = __builtin_amdgcn_wmma_f32_16x16x4_f32(false, Ls0[kk], false, nbi, (short)0, cre0, false, false);
            cim0 = __builtin_amdgcn_wmma_f32_16x16x4_f32(false, Lc0[kk], false, bi,  (short)0, cim0, false, false);
            cim0 = __builtin_amdgcn_wmma_f32_16x16x4_f32(false, Ls0[kk], false, br,  (short)0, cim0, false, false);
            cre1 = __builtin_amdgcn_wmma_f32_16x16x4_f32(false, Lc1[kk], false, br,  (short)0, cre1, false, false);
            cre1 = __builtin_amdgcn_wmma_f32_16x16x4_f32(false, Ls1[kk], false, nbi, (short)0, cre1, false, false);
            cim1 = __builtin_amdgcn_wmma_f32_16x16x4_f32(false, Lc1[kk], false, bi,  (short)0, cim1, false, false);
            cim1 = __builtin_amdgcn_wmma_f32_16x16x4_f32(false, Ls1[kk], false, br,  (short)0, cim1, false, false);
        }
        float s = scale[k];
#pragma unroll
        for (int r = 0; r < 8; ++r) {
            Iacc0[r] += s * (cre0[r]*cre0[r] + cim0[r]*cim0[r]);
            Iacc1[r] += s * (cre1[r]*cre1[r] + cim1[r]*cim1[r]);
        }
    }

    // dose^2 folding: x_out = I_f, x_out_min = 0.98^2 * I_d
    const float mul = set ? 0.9604f : 1.0f;
    float* dst = out + (set ? (size_t)4*SZ1 : (size_t)3*SZ1) + (size_t)n*PLANE;
#pragma unroll
    for (int r = 0; r < 8; ++r) {
        int row0 = ybase0 + r + 8*half;    // C/D layout: reg r -> M = r + 8*half
        int row1 = ybase1 + r + 8*half;
        dst[(size_t)row0*1024 + col] = mul * Iacc0[r];
        dst[(size_t)row1*1024 + col] = mul * Iacc1[r];
    }
}

// ---------------------------------------------------------------------------
// Kernel 5: elementwise resist maps + x_out_max from the stored intensities.
// ---------------------------------------------------------------------------
__global__ void __launch_bounds__(256) finalize_kernel(float* __restrict__ out)
{
    int i = blockIdx.x * 256 + threadIdx.x;
    if (i >= SZ1) return;
    float If   = out[(size_t)3*SZ1 + i];   // x_out (focus, dose 1)
    float Imin = out[(size_t)4*SZ1 + i];   // x_out_min = 0.9604*I_d
    float Imax = 1.0404f * If;             // 1.02^2 * I_f
    out[i]                 = sigmoidf_(50.0f * (If   - 0.225f));  // nominal
    out[(size_t)1*SZ1 + i] = sigmoidf_(50.0f * (Imin - 0.225f));  // inner
    out[(size_t)2*SZ1 + i] = sigmoidf_(50.0f * (Imax - 0.225f));  // outer
    out[(size_t)5*SZ1 + i] = Imax;                                 // x_out_max
}

extern "C" void kernel_launch(void* const* d_in, const int* in_sizes, int n_in,
                              void* d_out, int out_size, void* d_ws, size_t ws_size,
                              hipStream_t stream)
{
    (void)in_sizes; (void)n_in; (void)out_size; (void)ws_size;
    const float* mask = (const float*)d_in[0];
    const float* kfr  = (const float*)d_in[1];
    const float* kfi  = (const float*)d_in[2];
    const float* kdr  = (const float*)d_in[3];
    const float* kdi  = (const float*)d_in[4];
    const float* sf   = (const float*)d_in[5];
    const float* sd   = (const float*)d_in[6];
    float* out = (float*)d_out;
    float* ws  = (float*)d_ws;

    row_dft_kernel   <<<NIMG*HH,              128, 0, stream>>>(mask, ws);
    col_dft_kernel   <<<(NIMG*KDIM*KDIM+7)/8, 256, 0, stream>>>(ws);
    inv_stage1_kernel<<<2*NIMG*NK*KPAD,       256, 0, stream>>>(kfr, kfi, kdr, kdi, ws);
    wmma_intensity_kernel<<<1024,             256, 0, stream>>>(ws, sf, sd, out);
    finalize_kernel  <<<(SZ1+255)/256,        256, 0, stream>>>(out);
}